// StreamDIAM_89361089560603
// MI455X (gfx1250) — compile-verified
//
#include <hip/hip_runtime.h>
#include <math.h>

#define NN 50000
#define D 128
#define ATT_H 16

typedef float v2f __attribute__((ext_vector_type(2)));
typedef float v8f __attribute__((ext_vector_type(8)));

__device__ __forceinline__ void atomAddF(float* p, float v) {
    __hip_atomic_fetch_add(p, v, __ATOMIC_RELAXED, __HIP_MEMORY_SCOPE_AGENT);
}

// ---------------------------------------------------------------------------
// Kernel 1: zero the workspace (accumulators + degree counters)
// ---------------------------------------------------------------------------
__global__ void zero_ws_kernel(float* __restrict__ ws, int n) {
    int i = blockIdx.x * blockDim.x + threadIdx.x;
    int stride = gridDim.x * blockDim.x;
    for (; i < n; i += stride) ws[i] = 0.0f;
}

// ---------------------------------------------------------------------------
// Kernel 2: edge scatter.  One wave (32 lanes) per edge, grid-stride.
//   A_in[dst]  += x[src];  deg_in[dst]  += 1
//   A_out[src] += x[dst];  deg_out[src] += 1
// Lane l covers floats [4l, 4l+4) of the 128-wide feature row (float4 loads,
// 4 scalar fp32 atomics each).  All atomic traffic stays in L2 (working set
// << 192 MB).
// ---------------------------------------------------------------------------
__global__ void edge_scatter_kernel(const float* __restrict__ x,
                                    const int* __restrict__ ei, int E,
                                    float* __restrict__ A_in,
                                    float* __restrict__ A_out,
                                    float* __restrict__ deg_in,
                                    float* __restrict__ deg_out) {
    int gtid  = blockIdx.x * blockDim.x + threadIdx.x;
    int wave  = gtid >> 5;
    int lane  = gtid & 31;
    int nwave = (gridDim.x * blockDim.x) >> 5;

    for (int e = wave; e < E; e += nwave) {
        int s = ei[e];
        int d = ei[E + e];
        const float4 xs = *(const float4*)(x + (size_t)s * D + lane * 4);
        const float4 xd = *(const float4*)(x + (size_t)d * D + lane * 4);
        float* ain = A_in + (size_t)d * D + lane * 4;
        atomAddF(ain + 0, xs.x); atomAddF(ain + 1, xs.y);
        atomAddF(ain + 2, xs.z); atomAddF(ain + 3, xs.w);
        float* aout = A_out + (size_t)s * D + lane * 4;
        atomAddF(aout + 0, xd.x); atomAddF(aout + 1, xd.y);
        atomAddF(aout + 2, xd.z); atomAddF(aout + 3, xd.w);
        if (lane == 0) {
            atomAddF(deg_in + d, 1.0f);
            atomAddF(deg_out + s, 1.0f);
        }
    }
}

// ---------------------------------------------------------------------------
// Kernel 3: fused node transform + attention.
// Block = 256 threads = 8 waves, handles a tile of 16 nodes.
// Wave w computes output columns [16w, 16w+16) of four GEMMs via
// v_wmma_f32_16x16x4_f32 (K-loop of 32 steps over K=128):
//   acc_self = Xt @ W_self
//   acc_w2   = Xt @ W2            (W2 = W_disc rows 128..255)
//   acc_in   = A_in_t  @ (W1-W2)  (W1 = W_disc rows 0..127)
//   acc_out  = A_out_t @ (W1-W2)
// Epilogue reconstructs the three reps, runs tanh-MLP attention + softmax
// over the 3 reps, and writes the weighted sum.
// ---------------------------------------------------------------------------
__global__ __launch_bounds__(256) void node_fused_kernel(
        const float* __restrict__ x,
        const float* __restrict__ A_in, const float* __restrict__ A_out,
        const float* __restrict__ deg_in, const float* __restrict__ deg_out,
        const float* __restrict__ W_self, const float* __restrict__ b_self,
        const float* __restrict__ W_disc, const float* __restrict__ b_disc,
        const float* __restrict__ W_a1, const float* __restrict__ b_a1,
        const float* __restrict__ W_a2,
        float* __restrict__ out) {
    // Phase 1: sm = [ x_tile | A_in_tile | A_out_tile ], each 16x128.
    // Phase 2 (reused): sm = [ rep_self | rep_in | rep_out ].
    __shared__ float sm[3 * 16 * D];
    __shared__ float att_t[16 * 3 * ATT_H];
    __shared__ float att_s[16 * 3];
    __shared__ float degs[32];   // [0..15] deg_in, [16..31] deg_out

    const int tid  = threadIdx.x;
    const int base = blockIdx.x * 16;

    // ---- load tiles (float4, coalesced) ----
    for (int i = tid; i < (16 * D) / 4; i += 256) {   // 512 float4 per matrix
        int off  = i * 4;
        int row  = off >> 7;          // off / 128
        int col  = off & (D - 1);
        size_t g = (size_t)(base + row) * D + col;
        ((float4*)sm)[i]        = *(const float4*)(x     + g);
        ((float4*)sm)[512 + i]  = *(const float4*)(A_in  + g);
        ((float4*)sm)[1024 + i] = *(const float4*)(A_out + g);
    }
    if (tid < 16)       degs[tid] = deg_in[base + tid];
    else if (tid < 32)  degs[tid] = deg_out[base + tid - 16];
    __syncthreads();

    const int w    = tid >> 5;
    const int lane = tid & 31;
    const int half = lane >> 4;
    const int m    = lane & 15;
    const int ncol = w * 16 + m;      // output feature / B column for this lane

    v8f acc_self = {0,0,0,0,0,0,0,0};
    v8f acc_w2   = {0,0,0,0,0,0,0,0};
    v8f acc_in   = {0,0,0,0,0,0,0,0};
    v8f acc_out  = {0,0,0,0,0,0,0,0};

    // K loop: 32 steps x K=4.  A frag: lane holds M=m, K=kk..kk+1.
    //                          B frag: lane holds N=ncol, K=kk..kk+1.
    for (int k = 0; k < 32; ++k) {
        const int kk = 4 * k + 2 * half;
        v2f a_x = *(const v2f*)&sm[m * D + kk];
        v2f a_i = *(const v2f*)&sm[2048 + m * D + kk];
        v2f a_o = *(const v2f*)&sm[4096 + m * D + kk];
        v2f b_s, b_w1, b_w2, b_wd;
        b_s.x  = W_self[(size_t)kk * D + ncol];
        b_s.y  = W_self[(size_t)(kk + 1) * D + ncol];
        b_w1.x = W_disc[(size_t)kk * D + ncol];
        b_w1.y = W_disc[(size_t)(kk + 1) * D + ncol];
        b_w2.x = W_disc[(size_t)(128 + kk) * D + ncol];
        b_w2.y = W_disc[(size_t)(129 + kk) * D + ncol];
        b_wd.x = b_w1.x - b_w2.x;
        b_wd.y = b_w1.y - b_w2.y;

        acc_self = __builtin_amdgcn_wmma_f32_16x16x4_f32(
            false, a_x, false, b_s,  (short)0, acc_self, false, false);
        acc_w2   = __builtin_amdgcn_wmma_f32_16x16x4_f32(
            false, a_x, false, b_w2, (short)0, acc_w2,   false, false);
        acc_in   = __builtin_amdgcn_wmma_f32_16x16x4_f32(
            false, a_i, false, b_wd, (short)0, acc_in,   false, false);
        acc_out  = __builtin_amdgcn_wmma_f32_16x16x4_f32(
            false, a_o, false, b_wd, (short)0, acc_out,  false, false);
    }
    __syncthreads();   // everyone done reading the input tiles from LDS

    // ---- write the three reps back into the (reused) LDS tile ----
    // C/D layout: vgpr g -> row (g + 8*half), lane -> column m (of wave slab).
    const float bs = b_self[ncol];
    const float bd = b_disc[ncol];
#pragma unroll
    for (int g = 0; g < 8; ++g) {
        int row = g + 8 * half;
        float w2v = acc_w2[g] + bd;             // x@W2 + b_disc for this elem
        sm[row * D + ncol]        = acc_self[g] + bs;
        sm[2048 + row * D + ncol] = acc_in[g]  + degs[row]      * w2v;
        sm[4096 + row * D + ncol] = acc_out[g] + degs[16 + row] * w2v;
    }
    __syncthreads();

    // ---- attention MLP: t[row][rep][h] = tanh(rep_row . W_a1[:,h] + b_a1[h])
    for (int item = tid; item < 16 * 3 * ATT_H; item += 256) {
        int row = item / (3 * ATT_H);
        int rem = item - row * (3 * ATT_H);
        int rep = rem / ATT_H;
        int h   = rem - rep * ATT_H;
        const float* rp = &sm[rep * 2048 + row * D];
        float acc = b_a1[h];
        for (int c = 0; c < D; ++c) acc += rp[c] * W_a1[c * ATT_H + h];
        att_t[item] = tanhf(acc);
    }
    __syncthreads();

    // ---- logits + softmax over the 3 reps (one thread per row) ----
    if (tid < 16) {
        float lg[3];
        float mx = -1e30f;
#pragma unroll
        for (int rep = 0; rep < 3; ++rep) {
            float a = 0.0f;
            for (int h = 0; h < ATT_H; ++h)
                a += att_t[tid * (3 * ATT_H) + rep * ATT_H + h] * W_a2[h];
            lg[rep] = a;
            mx = fmaxf(mx, a);
        }
        float e0 = expf(lg[0] - mx), e1 = expf(lg[1] - mx), e2 = expf(lg[2] - mx);
        float inv = 1.0f / (e0 + e1 + e2);
        att_s[tid * 3 + 0] = e0 * inv;
        att_s[tid * 3 + 1] = e1 * inv;
        att_s[tid * 3 + 2] = e2 * inv;
    }
    __syncthreads();

    // ---- weighted sum of reps -> output ----
    for (int i = tid; i < 16 * D; i += 256) {
        int row = i >> 7;
        out[(size_t)(base + row) * D + (i & (D - 1))] =
            att_s[row * 3 + 0] * sm[i] +
            att_s[row * 3 + 1] * sm[2048 + i] +
            att_s[row * 3 + 2] * sm[4096 + i];
    }
}

// ---------------------------------------------------------------------------
extern "C" void kernel_launch(void* const* d_in, const int* in_sizes, int n_in,
                              void* d_out, int out_size, void* d_ws, size_t ws_size,
                              hipStream_t stream) {
    const float* x      = (const float*)d_in[0];
    const int*   ei     = (const int*)d_in[1];
    const float* W_self = (const float*)d_in[2];
    const float* b_self = (const float*)d_in[3];
    const float* W_disc = (const float*)d_in[4];
    const float* b_disc = (const float*)d_in[5];
    const float* W_a1   = (const float*)d_in[6];
    const float* b_a1   = (const float*)d_in[7];
    const float* W_a2   = (const float*)d_in[8];
    float* out = (float*)d_out;

    const int E = in_sizes[1] / 2;

    float* ws      = (float*)d_ws;
    float* A_in    = ws;
    float* A_out   = ws + (size_t)NN * D;
    float* deg_in  = ws + 2 * (size_t)NN * D;
    float* deg_out = deg_in + NN;
    const int zn = 2 * NN * D + 2 * NN;

    zero_ws_kernel<<<2048, 256, 0, stream>>>(ws, zn);
    edge_scatter_kernel<<<2048, 256, 0, stream>>>(x, ei, E, A_in, A_out,
                                                  deg_in, deg_out);
    node_fused_kernel<<<NN / 16, 256, 0, stream>>>(
        x, A_in, A_out, deg_in, deg_out,
        W_self, b_self, W_disc, b_disc, W_a1, b_a1, W_a2, out);
}